// SparseStructured_HopfieldCore_27075473834149
// MI455X (gfx1250) — compile-verified
//
#include <hip/hip_runtime.h>

// CDNA5 / gfx1250 top-k ("Hopfield") attention.
// B=2, L=S=2048, H=8, E=64, k=409, scale=1/8.
// Both GEMMs use exact-fp32 V_WMMA_F32_16X16X4_F32 so the top-k ordering
// matches the fp32 reference; threshold found by exact 4x8-bit radix select.

typedef __attribute__((ext_vector_type(2))) float v2f;
typedef __attribute__((ext_vector_type(8))) float v8f;

#define B_ 2
#define L_ 2048
#define S_ 2048
#define H_ 8
#define E_ 64
#define K_TOP 409
#define SCALE 0.125f
#define TM 16          // query rows per workgroup
#define NTHREADS 128   // 4 waves (wave32)

// Monotonic float->uint key: key(a) < key(b)  <=>  a < b
__device__ __forceinline__ unsigned int f2key(float f) {
  unsigned int u = __float_as_uint(f);
  return (u & 0x80000000u) ? ~u : (u | 0x80000000u);
}

__global__ __launch_bounds__(NTHREADS) void
topk_attn_kernel(const float* __restrict__ Q, const float* __restrict__ Km,
                 const float* __restrict__ Vm, float* __restrict__ Out) {
  // 128KB score/weight tile + 16KB histograms + row state: ~147.5KB of the
  // 320KB WGP LDS (2 workgroups per WGP).
  __shared__ __align__(16) float sc[TM][S_];
  __shared__ unsigned int hist[TM][256];
  __shared__ unsigned int prefixv[TM];
  __shared__ int remv[TM];
  __shared__ unsigned int eqcnt[TM];
  __shared__ float zrow[TM];

  const int tid = threadIdx.x;
  const int lane = tid & 31;
  const int wv = tid >> 5;     // wave id 0..3
  const int l16 = lane & 15;   // N (or M) position within 16
  const int lhi = lane >> 4;   // 0: K=0,1 half / M=0..7 ; 1: K=2,3 half / M=8..15

  const int blk = blockIdx.x;      // 0..2047
  const int bh = blk >> 7;         // (b,h)
  const int b = bh >> 3;
  const int h = bh & 7;
  const int rb = (blk & 127) * TM; // query-row base

  // ---------------- Phase 1: scores[16][2048] = Q_tile * K^T (fp32 WMMA) ---
  // A fragment (16x4 f32): lane l holds Q[rb + (l&15)][e + (l>>4)*2 .. +1]
  v2f aq[16];
  {
    const float* qrow = Q + (((size_t)b * L_ + rb + l16) * H_ + h) * E_ + lhi * 2;
#pragma unroll
    for (int es = 0; es < 16; ++es)
      aq[es] = *(const v2f*)(qrow + es * 4);
  }

  // each wave computes 512 consecutive score columns (32 chunks of 16)
  for (int c = 0; c < 32; ++c) {
    const int n0 = wv * 512 + c * 16;
    // B fragment (4x16 f32): lane l holds K[n0 + (l&15)][e + (l>>4)*2 .. +1]
    const float* krow = Km + (((size_t)b * S_ + n0 + l16) * H_ + h) * E_ + lhi * 2;
    v8f acc = {0.f, 0.f, 0.f, 0.f, 0.f, 0.f, 0.f, 0.f};
#pragma unroll
    for (int es = 0; es < 16; ++es) {
      v2f bk = *(const v2f*)(krow + es * 4);
      acc = __builtin_amdgcn_wmma_f32_16x16x4_f32(
          /*neg_a=*/false, aq[es], /*neg_b=*/false, bk,
          /*c_mod=*/(short)0, acc, /*reuse_a=*/false, /*reuse_b=*/false);
    }
    // D layout: lane l, vgpr v -> D[M = v + 8*(l>>4)][N = l&15]
#pragma unroll
    for (int v = 0; v < 8; ++v)
      sc[v + 8 * lhi][n0 + l16] = acc[v];
  }

  if (tid < TM) {
    prefixv[tid] = 0u;
    remv[tid] = K_TOP;
    eqcnt[tid] = 0u;
    zrow[tid] = 0.f;
  }
  __syncthreads();

  // ---------------- Phase 2: exact k-th-largest threshold (radix select) ---
  // 8 threads per row, 256 elements each.
  const int row = tid >> 3;
  const int cb = (tid & 7) * 256;
#pragma unroll 1
  for (int pass = 0; pass < 4; ++pass) {
    const int shift = 24 - 8 * pass;
    const unsigned int km = (pass == 0) ? 0u : (0xFFFFFFFFu << (shift + 8));
    for (int i = tid; i < TM * 256; i += NTHREADS)
      ((unsigned int*)hist)[i] = 0u;
    __syncthreads();
    const unsigned int pf = prefixv[row];
    for (int j = 0; j < 256; ++j) {
      unsigned int key = f2key(sc[row][cb + j]);
      if ((key & km) == pf)
        atomicAdd(&hist[row][(key >> shift) & 255u], 1u);
    }
    __syncthreads();
    if ((tid & 7) == 0) {  // row leader: walk bins from the top
      int r = remv[row];
      unsigned int pfn = prefixv[row];
      for (int bin = 255; bin >= 0; --bin) {
        int cnt = (int)hist[row][bin];
        if (r <= cnt) { pfn |= ((unsigned int)bin) << shift; break; }
        r -= cnt;
      }
      prefixv[row] = pfn;
      remv[row] = r;
    }
    __syncthreads();
  }

  const unsigned int Tthr = prefixv[row];          // key of k-th largest
  const unsigned int rEq = (unsigned int)remv[row]; // #ties at T to include

  // ---------------- Phase 3: weights in place; Z = sum of weights ----------
  // selected: w = exp(scale*s); not selected: w = exp(0) = 1 (scatter-on-zeros)
  float zs = 0.f;
  for (int j = 0; j < 256; ++j) {
    float s = sc[row][cb + j];
    unsigned int key = f2key(s);
    float w;
    if (key > Tthr) {
      w = __expf(SCALE * s);
    } else if (key == Tthr) {
      unsigned int c0 = atomicAdd(&eqcnt[row], 1u);
      w = (c0 < rEq) ? __expf(SCALE * s) : 1.0f;
    } else {
      w = 1.0f;
    }
    sc[row][cb + j] = w;
    zs += w;
  }
  atomicAdd(&zrow[row], zs);
  __syncthreads();

  // ---------------- Phase 4: out[16][64] = W[16][2048] * V (fp32 WMMA) -----
  // each wave owns one 16-wide slice of E
  const int e0 = wv * 16;
  v8f acc = {0.f, 0.f, 0.f, 0.f, 0.f, 0.f, 0.f, 0.f};
  const float* vbase = Vm + ((size_t)b * S_ * H_ + h) * E_ + e0 + l16;
#pragma unroll 4
  for (int ks = 0; ks < 512; ++ks) {
    const int s0 = ks * 4 + lhi * 2;
    // A fragment from LDS weights: lane l -> W[l&15][ks*4 + (l>>4)*2 .. +1]
    v2f aw = *(const v2f*)&sc[l16][s0];
    // B fragment: lane l -> V[s0][e0 + l&15], V[s0+1][e0 + l&15]
    v2f bv;
    bv.x = vbase[(size_t)s0 * (H_ * E_)];
    bv.y = vbase[(size_t)(s0 + 1) * (H_ * E_)];
    acc = __builtin_amdgcn_wmma_f32_16x16x4_f32(
        false, aw, false, bv, (short)0, acc, false, false);
  }
#pragma unroll
  for (int v = 0; v < 8; ++v) {
    const int m = v + 8 * lhi;
    const float o = acc[v] / zrow[m];
    Out[(((size_t)b * L_ + rb + m) * H_ + h) * E_ + e0 + l16] = o;
  }
}

extern "C" void kernel_launch(void* const* d_in, const int* in_sizes, int n_in,
                              void* d_out, int out_size, void* d_ws, size_t ws_size,
                              hipStream_t stream) {
  (void)in_sizes; (void)n_in; (void)d_ws; (void)ws_size; (void)out_size;
  const float* Q = (const float*)d_in[0];
  const float* K = (const float*)d_in[1];
  const float* V = (const float*)d_in[2];
  float* O = (float*)d_out;
  dim3 grid(B_ * H_ * (L_ / TM));  // 2048 workgroups
  dim3 block(NTHREADS);            // 4 wave32 waves
  hipLaunchKernelGGL(topk_attn_kernel, grid, block, 0, stream, Q, K, V, O);
}